// MyAttention_34342558499027
// MI455X (gfx1250) — compile-verified
//
#include <hip/hip_runtime.h>
#include <cstdint>

typedef __attribute__((ext_vector_type(2))) float v2f;
typedef __attribute__((ext_vector_type(4))) float v4f;
typedef __attribute__((ext_vector_type(8))) float v8f;

#define B_ 32
#define T_ 1024
#define D_ 512
#define U_ 64
#define SR_ 10.0f
#define EPS_ 1e-7f

__device__ __forceinline__ v8f wmma_f32(v2f a, v2f b, v8f c) {
  // D = A(16x4) * B(4x16) + C, fp32 throughout
  return __builtin_amdgcn_wmma_f32_16x16x4_f32(false, a, false, b, (short)0, c,
                                               false, false);
}

// ---------------------------------------------------------------------------
// Stage 1: M1[b] = x2^T@w2 + x1^T@w3 + w1 ; M2[b] = x1^T@w2 + x2^T@w3 + w1
// grid (D/16, B), block 32 (one wave). Each wave owns a 16 x 64 tile of both.
// ---------------------------------------------------------------------------
__global__ __launch_bounds__(32) void k_stage1(
    const float* __restrict__ x1, const float* __restrict__ x2,
    const float* __restrict__ w1, const float* __restrict__ w2,
    const float* __restrict__ w3, float* __restrict__ M) {
  const int d_base = blockIdx.x * 16;
  const int b = blockIdx.y;
  const int lane = threadIdx.x;
  const int half = lane >> 4;
  const int lm = lane & 15;

  const float* x1b = x1 + (size_t)b * T_ * D_;
  const float* x2b = x2 + (size_t)b * T_ * D_;

  v8f c1[4] = {};
  v8f c2[4] = {};

  for (int t0 = 0; t0 < T_; t0 += 4) {
    const int tk = t0 + half * 2;  // K rows for this half-wave (v0=tk, v1=tk+1)
    const float* p1 = x1b + (size_t)tk * D_ + d_base + lm;
    const float* p2 = x2b + (size_t)tk * D_ + d_base + lm;
    v2f a1, a2;  // A = x^T : A[m=d][k=t]
    a1.x = p1[0];
    a1.y = p1[D_];
    a2.x = p2[0];
    a2.y = p2[D_];
#pragma unroll
    for (int ut = 0; ut < 4; ++ut) {
      const float* q2 = w2 + (size_t)tk * U_ + ut * 16 + lm;
      const float* q3 = w3 + (size_t)tk * U_ + ut * 16 + lm;
      v2f b2, b3;  // B[k=t][n=u]
      b2.x = q2[0];
      b2.y = q2[U_];
      b3.x = q3[0];
      b3.y = q3[U_];
      c1[ut] = wmma_f32(a2, b2, c1[ut]);  // x2^T @ w2
      c1[ut] = wmma_f32(a1, b3, c1[ut]);  // x1^T @ w3
      c2[ut] = wmma_f32(a1, b2, c2[ut]);  // x1^T @ w2
      c2[ut] = wmma_f32(a2, b3, c2[ut]);  // x2^T @ w3
    }
  }

  float* M1 = M + (size_t)b * D_ * U_;
  float* M2 = M + (size_t)(B_ + b) * D_ * U_;
#pragma unroll
  for (int ut = 0; ut < 4; ++ut) {
#pragma unroll
    for (int i = 0; i < 8; ++i) {
      const int m = d_base + i + 8 * half;  // C layout: M = i + 8*half
      const int n = ut * 16 + lm;           // N = lane & 15
      const float w1v = w1[(size_t)m * U_ + n];
      M1[(size_t)m * U_ + n] = c1[ut][i] + w1v;
      M2[(size_t)m * U_ + n] = c2[ut][i] + w1v;
    }
  }
}

// ---------------------------------------------------------------------------
// Stage 2: h = tanh(x_r @ M_r) (16 x 64 tile per wave), e = SR * (h @ we)
// grid (T/16, B, 2), block 32.
// ---------------------------------------------------------------------------
__global__ __launch_bounds__(32) void k_stage2(
    const float* __restrict__ x1, const float* __restrict__ x2,
    const float* __restrict__ M, const float* __restrict__ we,
    float* __restrict__ E) {
  const int t_base = blockIdx.x * 16;
  const int b = blockIdx.y;
  const int r = blockIdx.z;
  const int lane = threadIdx.x;
  const int half = lane >> 4;
  const int lm = lane & 15;

  const float* xa = (r == 0 ? x1 : x2) + (size_t)b * T_ * D_;
  const float* Mb = M + (size_t)(r * B_ + b) * D_ * U_;

  v8f c[4] = {};

  for (int d0 = 0; d0 < D_; d0 += 4) {
    const int dk = d0 + half * 2;
    const float* p = xa + (size_t)(t_base + lm) * D_ + dk;
    v2f a;  // A[m=t][k=d]; two consecutive floats -> single b64 load
    a.x = p[0];
    a.y = p[1];
#pragma unroll
    for (int ut = 0; ut < 4; ++ut) {
      const float* q = Mb + (size_t)dk * U_ + ut * 16 + lm;
      v2f bm;  // B[k=d][n=u]
      bm.x = q[0];
      bm.y = q[U_];
      c[ut] = wmma_f32(a, bm, c[ut]);
    }
  }

  float wev[4];
#pragma unroll
  for (int ut = 0; ut < 4; ++ut) wev[ut] = we[ut * 16 + lm];

  float* Eb = E + (size_t)(r * B_ + b) * T_;
#pragma unroll
  for (int i = 0; i < 8; ++i) {
    float v = 0.0f;
#pragma unroll
    for (int ut = 0; ut < 4; ++ut) v += tanhf(c[ut][i]) * wev[ut];
    // reduce across the 16 lanes of each half-wave (xor masks stay in-half)
    v += __shfl_xor(v, 1, 32);
    v += __shfl_xor(v, 2, 32);
    v += __shfl_xor(v, 4, 32);
    v += __shfl_xor(v, 8, 32);
    if (lm == 0) Eb[t_base + i + 8 * half] = SR_ * v;
  }
}

// ---------------------------------------------------------------------------
// Stage 3: S[r,b] = sum_t exp(e[r,b,t]). grid 2*B, block 256.
// ---------------------------------------------------------------------------
__global__ __launch_bounds__(256) void k_sum(const float* __restrict__ E,
                                             float* __restrict__ S) {
  const int rb = blockIdx.x;
  const float* Eb = E + (size_t)rb * T_;
  __shared__ float red[256];
  float s = 0.0f;
  for (int t = threadIdx.x; t < T_; t += 256) s += expf(Eb[t]);
  red[threadIdx.x] = s;
  __syncthreads();
  for (int off = 128; off > 0; off >>= 1) {
    if ((int)threadIdx.x < off) red[threadIdx.x] += red[threadIdx.x + off];
    __syncthreads();
  }
  if (threadIdx.x == 0) S[rb] = red[0];
}

// ---------------------------------------------------------------------------
// Stage 4: out[r,b,t,:] = x_r[b,t,:] * exp(e)/(S+eps). float4 vectorized.
// ---------------------------------------------------------------------------
__global__ __launch_bounds__(256) void k_scale(
    const float* __restrict__ x1, const float* __restrict__ x2,
    const float* __restrict__ E, const float* __restrict__ S,
    float* __restrict__ out) {
  const size_t idx = (size_t)blockIdx.x * blockDim.x + threadIdx.x;  // float4 idx
  const int d4 = (int)(idx & (D_ / 4 - 1));
  const size_t lin = idx >> 7;  // r*B*T + b*T + t
  const int t = (int)(lin & (T_ - 1));
  const int b = (int)((lin >> 10) & (B_ - 1));
  const int r = (int)(lin >> 15);

  const float w = expf(E[lin]) / (S[(size_t)r * B_ + b] + EPS_);
  const v4f* xp =
      (const v4f*)((r == 0 ? x1 : x2) + ((size_t)b * T_ + t) * D_) + d4;
  v4f v = *xp;
  v.x *= w;
  v.y *= w;
  v.z *= w;
  v.w *= w;
  ((v4f*)out)[idx] = v;
}

extern "C" void kernel_launch(void* const* d_in, const int* in_sizes, int n_in,
                              void* d_out, int out_size, void* d_ws,
                              size_t ws_size, hipStream_t stream) {
  (void)in_sizes;
  (void)n_in;
  (void)out_size;
  (void)ws_size;
  const float* x1 = (const float*)d_in[0];
  const float* x2 = (const float*)d_in[1];
  const float* w1 = (const float*)d_in[2];
  const float* w2 = (const float*)d_in[3];
  const float* w3 = (const float*)d_in[4];
  const float* we = (const float*)d_in[5];
  float* out = (float*)d_out;

  float* M = (float*)d_ws;                      // 2*B*D*U floats = 8 MB
  float* E = M + (size_t)2 * B_ * D_ * U_;      // 2*B*T floats
  float* S = E + (size_t)2 * B_ * T_;           // 2*B floats

  k_stage1<<<dim3(D_ / 16, B_), 32, 0, stream>>>(x1, x2, w1, w2, w3, M);
  k_stage2<<<dim3(T_ / 16, B_, 2), 32, 0, stream>>>(x1, x2, M, we, E);
  k_sum<<<2 * B_, 256, 0, stream>>>(E, S);
  const size_t n4 = (size_t)2 * B_ * T_ * D_ / 4;
  k_scale<<<(unsigned)(n4 / 256), 256, 0, stream>>>(x1, x2, E, S, out);
}